// EdgeEncoding_17935783428481
// MI455X (gfx1250) — compile-verified
//
#include <hip/hip_runtime.h>

// MI455X / gfx1250, wave32. Fused edge-encoding kernel:
//   per-source LDS segment-sum (ds_add_f32) + V_WMMA_F32_16X16X4_F32 for the
//   [1M,4]x[4,16] linear layer, transposed coalesced store to out[16,1024,1024].

typedef float v2f __attribute__((ext_vector_type(2)));
typedef float v8f __attribute__((ext_vector_type(8)));

#define GN   1024            // N nodes
#define GH   16              // heads
#define GNN  (GN * GN)

__launch_bounds__(256)
__global__ void edge_encoding_kernel(const float* __restrict__ edge_attr,   // [N,4]
                                     const float* __restrict__ W,           // [16,4]
                                     const float* __restrict__ b,           // [16]
                                     const int*   __restrict__ pair_id,     // [P]
                                     const int*   __restrict__ node_id,     // [P]
                                     const int*   __restrict__ path_len,    // [N*N]
                                     float*       __restrict__ out,         // [16,N,N]
                                     int P)
{
    __shared__ __align__(16) float accum[GN * 4];   // 16 KB segment sums for this source
    __shared__ __align__(16) float ea[GN * 4];      // 16 KB cached edge_attr
    __shared__ __align__(16) float Ws[GH * 4];
    __shared__ __align__(16) float bs[GH];

    const int s   = blockIdx.x;      // source node
    const int tid = threadIdx.x;

    // ---- init LDS: zero accumulator, cache edge_attr / W / b ----
    for (int i = tid; i < GN * 4; i += 256) {
        accum[i] = 0.0f;
        ea[i]    = edge_attr[i];
    }
    if (tid < GH * 4) Ws[tid] = W[tid];
    if (tid < GH)     bs[tid] = b[tid];
    __syncthreads();

    // ---- binary search the contiguous entry range of source s ----
    // pair_id for source s lies in [s*GN, (s+1)*GN); sources are contiguous,
    // so (pair_id[m] >= s*GN) is monotone over the whole array.
    int lo, hi;
    {
        int target = s * GN;
        int l = 0, r = P;
        while (l < r) { int m = (l + r) >> 1; if (pair_id[m] < target) l = m + 1; else r = m; }
        lo = l;
        target = (s + 1) * GN;
        r = P;
        while (l < r) { int m = (l + r) >> 1; if (pair_id[m] < target) l = m + 1; else r = m; }
        hi = l;
    }

    // ---- phase 1: segment sum into LDS via ds_add_f32 atomics ----
    for (int e = lo + tid; e < hi; e += 256) {
        int j = pair_id[e] - s * GN;            // destination column (0..1023)
        int n = node_id[e];
        float4 v = *(const float4*)(&ea[n * 4]);
        atomicAdd(&accum[j * 4 + 0], v.x);
        atomicAdd(&accum[j * 4 + 1], v.y);
        atomicAdd(&accum[j * 4 + 2], v.z);
        atomicAdd(&accum[j * 4 + 3], v.w);
    }
    __syncthreads();

    // ---- phase 2: D = A(=W, 16x4) x B(=sums/L, 4x16) + C(=b masked) ----
    // wave32 lane mapping for V_WMMA_F32_16X16X4_F32:
    //   A: VGPR0 = K (0 | 2) split across lane halves, VGPR1 = K (1 | 3), M = lane&15
    //   B: same with N = lane&15
    //   D: VGPR r -> h = r + 8*(lane>=16), pair column = lane&15
    const int lane   = tid & 31;
    const int wave   = tid >> 5;
    const int l15    = lane & 15;
    const int half   = lane >> 4;        // 0: lanes 0-15, 1: lanes 16-31
    const int k0     = half * 2;

    v2f A;
    A.x = Ws[l15 * 4 + k0];
    A.y = Ws[l15 * 4 + k0 + 1];

    // loop-invariant bias fragment: two b128 LDS loads, no predication
    const float4 blo = *(const float4*)(&bs[half * 8]);
    const float4 bhi = *(const float4*)(&bs[half * 8 + 4]);
    float bvals[8] = { blo.x, blo.y, blo.z, blo.w, bhi.x, bhi.y, bhi.z, bhi.w };

    #pragma unroll
    for (int i = 0; i < 8; ++i) {
        const int tile = wave * 8 + i;       // 64 tiles of 16 pairs per block
        const int p0   = tile * 16;
        const int pidx = p0 + l15;           // this lane's pair column

        const int   L    = path_len[s * GN + pidx];
        // fast 1-ulp reciprocal; L is a small exact integer (accum==0 when L==0)
        const float rL   = __builtin_amdgcn_rcpf(fmaxf((float)L, 1.0f));
        const float mask = (L > 0) ? 1.0f : 0.0f;   // single v_cndmask

        v2f B;
        B.x = accum[pidx * 4 + k0]     * rL;
        B.y = accum[pidx * 4 + k0 + 1] * rL;

        v8f C;
        #pragma unroll
        for (int r = 0; r < 8; ++r)
            C[r] = bvals[r] * mask;          // plain VALU, dual-issueable

        v8f D = __builtin_amdgcn_wmma_f32_16x16x4_f32(
                    /*neg_a=*/false, A, /*neg_b=*/false, B,
                    /*c_mod=*/(short)0, C, /*reuse_a=*/false, /*reuse_b=*/false);

        // 32-bit offsets -> saddr-form global_store, no 64-bit carry chains
        const unsigned base = (unsigned)(s * GN + pidx);
        #pragma unroll
        for (int r = 0; r < 8; ++r) {
            const unsigned h = (unsigned)(r + half * 8);
            out[h * (unsigned)GNN + base] = D[r];   // 64B-coalesced per lane half
        }
    }
}

extern "C" void kernel_launch(void* const* d_in, const int* in_sizes, int n_in,
                              void* d_out, int out_size, void* d_ws, size_t ws_size,
                              hipStream_t stream) {
    // setup_inputs order: x, edge_attr, W, b, pair_id, node_id, path_len
    const float* edge_attr = (const float*)d_in[1];
    const float* W         = (const float*)d_in[2];
    const float* b         = (const float*)d_in[3];
    const int*   pair_id   = (const int*)d_in[4];
    const int*   node_id   = (const int*)d_in[5];
    const int*   path_len  = (const int*)d_in[6];
    float*       out       = (float*)d_out;
    const int    P         = in_sizes[4];

    edge_encoding_kernel<<<dim3(GN), dim3(256), 0, stream>>>(
        edge_attr, W, b, pair_id, node_id, path_len, out, P);
}